// Synthesiser3D_88098369175865
// MI455X (gfx1250) — compile-verified
//
#include <hip/hip_runtime.h>
#include <math.h>

// Problem constants (from reference): source (2,32,256,256) f32, nnf (2,3,256,256) f32
#define BS 2
#define CH 32
#define H  256
#define W  256
#define HW (H * W)

// Detect the gfx1250 async LDS->global store builtin (ASYNCcnt path).
#if defined(__has_builtin)
#  if __has_builtin(__builtin_amdgcn_global_store_async_from_lds_b128)
#    define HAVE_ASYNC_STORE 1
#  endif
#endif
#ifndef HAVE_ASYNC_STORE
#  define HAVE_ASYNC_STORE 0
#endif

// Pointer types expected by the async builtin (from hipcc's diagnostic:
// parameter is "int __vector_size__(4*sizeof(int)) __device__ *").
typedef int v4i __attribute__((__vector_size__(4 * sizeof(int))));
typedef __attribute__((address_space(1))) v4i* global_v4i_ptr;
typedef __attribute__((address_space(3))) v4i* shared_v4i_ptr;

// ---------------------------------------------------------------------------
// Pass 1: NCHW -> NHWC staging so each gathered pixel is ONE 128B cacheline
// (32 channels x 4B). Streaming 16.7MB copy; writes fully coalesced.
// ---------------------------------------------------------------------------
__global__ __launch_bounds__(256) void k_nchw_to_nhwc(const float* __restrict__ src,
                                                      float* __restrict__ dst) {
  unsigned i = blockIdx.x * 256u + threadIdx.x;   // NHWC linear index, c fastest
  unsigned c = i & 31u;
  unsigned x = (i >> 5) & 255u;
  unsigned y = (i >> 13) & 255u;
  unsigned b = i >> 21;
  dst[i] = src[((size_t)(b * CH + c)) * HW + (size_t)y * W + x];
}

// ---------------------------------------------------------------------------
// Pass 2: one wave32 per output pixel, lane == channel.
//   * lanes 0..24 each compute one rotated patch position's flat offset
//   * v_readlane broadcasts the uniform offset -> 25 fully-coalesced
//     single-cacheline gathers (NHWC) accumulated in f32
//   * per-block result tile (32ch x 8px) transposed through LDS, then drained
//     with global_store_async_from_lds_b128 (ASYNCcnt; implicit wait at ENDPGM)
// ---------------------------------------------------------------------------
template <bool USE_NHWC>
__global__ __launch_bounds__(256) void k_synth(const float* __restrict__ src,
                                               const float* __restrict__ nnf,
                                               float* __restrict__ out) {
  __shared__ __align__(128) float tile[CH * 8];   // [channel][x_in_block]

  const int tid  = threadIdx.x;
  const int lane = tid & 31;       // channel index
  const int wv   = tid >> 5;       // 0..7 -> which pixel of the 8-pixel block

  // blockIdx -> (b, y, xg): grid = BS * H * (W/8) = 16384
  const unsigned bid = blockIdx.x;
  const int xg = bid & 31;                 // 32 groups of 8 along x
  const int y  = (bid >> 5) & 255;
  const int b  = (int)(bid >> 13);
  const int x  = (xg << 3) + wv;

  // Uniform (per-wave) nnf reads for this pixel.
  const float* nb = nnf + (size_t)b * 3 * HW + (size_t)y * W + x;
  const float ci  = nb[0];        // row coordinate, in [0, H-1]
  const float cj  = nb[HW];       // col coordinate, in [0, W-1]
  const float ang = nb[2 * HW];   // angle in [0,1)

  float sv, cv;
  sincosf(ang * 3.14159265358979323846f, &sv, &cv);   // accurate libm path

  // Each lane p (=lane) owns patch element (i = p/5, j = p%5); di = i-2, dj = j-2.
  // Reference: pi_i = dj, pi_j = di;  iR = dj*s - di*c;  jR = dj*c - di*s.
  // Lanes 25..31 compute harmless clamped values that are never consumed.
  const float di = (float)(lane / 5) - 2.0f;
  const float dj = (float)(lane % 5) - 2.0f;
  const float iR = dj * sv - di * cv;
  const float jR = dj * cv - di * sv;
  float pif = fminf(fmaxf(ci + iR, 0.0f), (float)(H - 1));
  float pjf = fminf(fmaxf(cj + jR, 0.0f), (float)(W - 1));
  const int  flat = (int)pif * W + (int)pjf;             // clamp then trunc == ref
  const int  eofs = USE_NHWC ? flat * CH : flat;

  const float* base = USE_NHWC ? (src + (size_t)b * HW * CH)
                               : (src + ((size_t)b * CH + lane) * HW);

  float acc = 0.0f;
#pragma unroll
  for (int p = 0; p < 25; ++p) {
    const int o = __builtin_amdgcn_readlane(eofs, p);    // uniform broadcast (SGPR)
    acc += USE_NHWC ? base[o + lane]                     // one 128B line / load
                    : base[o];                           // fallback: strided
  }

  // Transpose through LDS so stores go out channel-row contiguous.
  tile[lane * 8 + wv] = acc;
  __syncthreads();

  float* outp = out + (size_t)b * CH * HW + (size_t)y * W + (xg << 3);

#if HAVE_ASYNC_STORE
  // Drain the 1KB tile with 2 async b128 instructions from wave 0.
  // Each chunk t: channel cc = t>>1, 16B half = t&1. Global + LDS both 16B aligned.
  // S_ENDPGM's implicit wait-idle covers ASYNCcnt before LDS dealloc.
  if (wv == 0) {
#pragma unroll
    for (int i = 0; i < 2; ++i) {
      const int t    = i * 32 + lane;
      const int cc   = t >> 1;
      const int half = t & 1;
      float* g = outp + (size_t)cc * HW + half * 4;
      const float* l = &tile[cc * 8 + half * 4];
      __builtin_amdgcn_global_store_async_from_lds_b128(
          (global_v4i_ptr)g,
          (shared_v4i_ptr)l,
          0, 0);
    }
  }
#else
  {
    const int cc = tid >> 3;     // channel
    const int xi = tid & 7;      // x within block -> 32B-contiguous per 8 threads
    outp[(size_t)cc * HW + xi] = tile[cc * 8 + xi];
  }
#endif
}

// ---------------------------------------------------------------------------
extern "C" void kernel_launch(void* const* d_in, const int* in_sizes, int n_in,
                              void* d_out, int out_size, void* d_ws, size_t ws_size,
                              hipStream_t stream) {
  const float* src = (const float*)d_in[0];   // (2,32,256,256)
  const float* nnf = (const float*)d_in[1];   // (2,3,256,256)
  float* out = (float*)d_out;                 // (2,32,256,256)

  const size_t nhwcBytes = (size_t)BS * CH * HW * sizeof(float);  // 16 MB
  if (d_ws && ws_size >= nhwcBytes) {
    float* nhwc = (float*)d_ws;
    k_nchw_to_nhwc<<<(BS * CH * HW) / 256, 256, 0, stream>>>(src, nhwc);
    k_synth<true><<<BS * H * (W / 8), 256, 0, stream>>>(nhwc, nnf, out);
  } else {
    // Workspace too small: gather straight from NCHW (correct, less coalesced).
    k_synth<false><<<BS * H * (W / 8), 256, 0, stream>>>(src, nnf, out);
  }
}